// RGraphConvolutionLayer_68831145885834
// MI455X (gfx1250) — compile-verified
//
#include <hip/hip_runtime.h>

// ---------------------------------------------------------------------------
// R-GCN layer, reordered as: for r in 0..R-1: y = x @ W_r (WMMA f32 GEMM),
// then out[row_e] += val_e * y[col_e] for all edges of relation r.
// Sequential relations keep {x, y, out} (~153MB) resident in the 192MB L2,
// so edge gathers and float atomics are L2 traffic, not HBM RMW.
// ---------------------------------------------------------------------------

typedef __attribute__((ext_vector_type(2))) float v2f;
typedef __attribute__((ext_vector_type(8))) float v8f;

#define D_DIM 128
#define O_DIM 128
// W is staged in LDS K-pair interleaved: element (k,n) -> sW[(k/2)*PAIR_STRIDE
// + 2n + (k&1)].  A lane's B fragment {W[k][n], W[k+1][n]} is then a single
// aligned ds_load_b64 into an even VGPR pair (no v_mov repacking).
// PAIR_STRIDE = 288 dwords: 288 % 64 == 32, so lanes 16-31 (pair p+1) hit
// banks 32..63 while lanes 0-15 (pair p) hit banks 0..31 -> conflict-free.
#define PAIR_STRIDE 288
#define N_PAIRS (D_DIM / 2)

// ---------------------------------------------------------------------------
// Zero-fill d_out (we accumulate atomically into it).
// ---------------------------------------------------------------------------
__global__ void rgcn_zero_kernel(float4* __restrict__ p, int n4) {
    int i = blockIdx.x * blockDim.x + threadIdx.x;
    if (i < n4) p[i] = make_float4(0.f, 0.f, 0.f, 0.f);
}

// ---------------------------------------------------------------------------
// y[m][o] = sum_d x[m][d] * Wr[d][o],  D = O = 128, fp32 WMMA 16x16x4.
// Block = 256 threads = 8 waves; wave w computes rows [ (blk*8+w)*16, +16 ).
// Wr staged (pair-interleaved) in LDS, shared by all 8 waves.
// ---------------------------------------------------------------------------
__global__ void __launch_bounds__(256)
rgcn_gemm_xw_kernel(const float* __restrict__ x,
                    const float* __restrict__ Wr,
                    float* __restrict__ y,
                    int n_rows) {
    __shared__ float sW[N_PAIRS * PAIR_STRIDE];   // 72 KB

    const int tid = threadIdx.x;

    // Cooperative load of Wr (128x128 f32) into pair-interleaved LDS.
    for (int g = tid; g < D_DIM * (O_DIM / 4); g += 256) {
        int d  = g >> 5;          // W row (K) 0..127
        int cq = g & 31;          // float4 within row
        float4 w4 = *(const float4*)(Wr + (size_t)d * O_DIM + cq * 4);
        float* b = &sW[(d >> 1) * PAIR_STRIDE + cq * 8 + (d & 1)];
        b[0] = w4.x; b[2] = w4.y; b[4] = w4.z; b[6] = w4.w;
    }
    __syncthreads();

    const int wave = tid >> 5;
    const int lane = tid & 31;
    const int m0   = (blockIdx.x * 8 + wave) * 16;
    if (m0 >= n_rows) return;          // wave-uniform: EXEC stays all-ones

    const int mrow  = lane & 15;       // A fragment: M = m0 + mrow
    const int pl    = lane >> 4;       // 0: K=k0,k0+1   1: K=k0+2,k0+3
    const int khalf = pl * 2;
    const int ncol  = lane & 15;       // B/C fragment column within 16-wide tile

    v8f acc[8] = {};                   // 16 x 128 fp32 output strip

    const float* xrow = x + (size_t)(m0 + mrow) * D_DIM;

    for (int k0 = 0; k0 < D_DIM; k0 += 4) {
        v2f a = *(const v2f*)(xrow + k0 + khalf);                 // b64 load
        const float* sWp = &sW[((k0 >> 1) + pl) * PAIR_STRIDE + ncol * 2];
#pragma unroll
        for (int t = 0; t < 8; ++t) {
            v2f b = *(const v2f*)(sWp + t * 32);                  // ds_load_b64
            acc[t] = __builtin_amdgcn_wmma_f32_16x16x4_f32(
                /*neg_a=*/false, a, /*neg_b=*/false, b,
                /*c_mod=*/(short)0, acc[t],
                /*reuse_a=*/false, /*reuse_b=*/false);
        }
    }

    // C/D layout: VGPR v -> M = m0 + v + (lane>=16 ? 8 : 0), N = t*16 + (lane&15)
    const int mbase = m0 + (pl << 3);
#pragma unroll
    for (int t = 0; t < 8; ++t) {
#pragma unroll
        for (int v = 0; v < 8; ++v) {
            y[(size_t)(mbase + v) * O_DIM + t * 16 + ncol] = acc[t][v];
        }
    }
}

// ---------------------------------------------------------------------------
// out[row_e][:] += val_e * y[col_e][:]   (one wave per edge, float4 per lane)
// out fits in L2 (51MB) -> atomics resolve in-cache.
// ---------------------------------------------------------------------------
__global__ void __launch_bounds__(256)
rgcn_scatter_kernel(const int* __restrict__ erow,
                    const int* __restrict__ ecol,
                    const float* __restrict__ eval,
                    const float* __restrict__ y,
                    float* __restrict__ out,
                    int n_edges) {
    const int wave = (blockIdx.x * blockDim.x + threadIdx.x) >> 5;
    const int lane = threadIdx.x & 31;
    if (wave >= n_edges) return;

    const int   r = erow[wave];
    const int   c = ecol[wave];
    const float v = eval[wave];

    float4 f = *(const float4*)(y + (size_t)c * O_DIM + lane * 4);
    float* dst = out + (size_t)r * O_DIM + lane * 4;
    atomicAdd(dst + 0, v * f.x);
    atomicAdd(dst + 1, v * f.y);
    atomicAdd(dst + 2, v * f.z);
    atomicAdd(dst + 3, v * f.w);
}

// ---------------------------------------------------------------------------
// Launch
// ---------------------------------------------------------------------------
extern "C" void kernel_launch(void* const* d_in, const int* in_sizes, int n_in,
                              void* d_out, int out_size, void* d_ws, size_t ws_size,
                              hipStream_t stream) {
    const float* x    = (const float*)d_in[0];   // [N, 128]
    const int*   erow = (const int*)  d_in[1];   // [R, E]
    const int*   ecol = (const int*)  d_in[2];   // [R, E]
    const float* eval = (const float*)d_in[3];   // [R, E]
    const float* W    = (const float*)d_in[4];   // [R*128, 128]
    float*       out  = (float*)d_out;           // [N, 128]
    float*       y    = (float*)d_ws;            // [N, 128] scratch (51.2 MB)

    const int R = 4;
    const int N = in_sizes[0] / D_DIM;
    const int E = in_sizes[1] / R;

    // Zero the accumulator output.
    {
        int n4 = (N * O_DIM) / 4;
        int blocks = (n4 + 255) / 256;
        rgcn_zero_kernel<<<blocks, 256, 0, stream>>>((float4*)out, n4);
    }

    const int wave_tiles = (N + 15) / 16;           // 16-row strips
    const int gemm_blocks = (wave_tiles + 7) / 8;   // 8 strips per block
    const int scat_blocks = (E + 7) / 8;            // 8 edges (waves) per block

    for (int r = 0; r < R; ++r) {
        rgcn_gemm_xw_kernel<<<gemm_blocks, 256, 0, stream>>>(
            x, W + (size_t)r * D_DIM * O_DIM, y, N);
        rgcn_scatter_kernel<<<scat_blocks, 256, 0, stream>>>(
            erow + (size_t)r * E, ecol + (size_t)r * E, eval + (size_t)r * E,
            y, out, E);
    }
}